// FWHT_26843545600266
// MI455X (gfx1250) — compile-verified
//
#include <hip/hip_runtime.h>

// FWHT-8192 per row via H_8192 = H16 (x) H16 (x) H16 (x) H2.
// - Row staged into LDS by the Tensor Data Mover (TENSOR_LOAD_TO_LDS) when the
//   builtin is available (fallback: coalesced B128 copy).
// - Three in-place WMMA passes (V_WMMA_F32_16X16X4_F32, exact f32 math).
// - H2 butterfly fused into non-temporal coalesced global stores.
// One block = one row = 32KB LDS; 8 waves; memory-bound (~512MB total traffic).

typedef float v2f __attribute__((ext_vector_type(2)));
typedef float v8f __attribute__((ext_vector_type(8)));
typedef unsigned int v4u __attribute__((ext_vector_type(4)));
typedef int v4i __attribute__((ext_vector_type(4)));
typedef int v8i __attribute__((ext_vector_type(8)));

__device__ __forceinline__ float hsign(int m, int k) {
    return (__popc(m & k) & 1) ? -1.0f : 1.0f;
}

__global__ __launch_bounds__(256, 2)
void fwht8192_wmma_kernel(const float* __restrict__ in, float* __restrict__ out) {
    __shared__ __align__(16) float lds[8192];

    const int tid  = threadIdx.x;       // 0..255
    const int lane = tid & 31;          // wave32 lane
    const int wave = tid >> 5;          // 0..7
    const int row  = blockIdx.x;

    // ---- Stage one full row (8192 f32 = 32KB) into LDS ----
#if __has_builtin(__builtin_amdgcn_tensor_load_to_lds) && __has_builtin(__builtin_amdgcn_s_wait_tensorcnt)
    if (wave == 0) {
        const unsigned long long gaddr =
            (unsigned long long)(const void*)(in + (size_t)row * 8192);
        // D# group 0: count=1 (valid user desc), lds_addr=0, global_addr, type=2.
        v4u g0;
        g0[0] = 1u;
        g0[1] = 0u;                                        // lds byte offset
        g0[2] = (unsigned)(gaddr & 0xFFFFFFFFu);           // global_addr[31:0]
        g0[3] = (unsigned)((gaddr >> 32) & 0x1FFFFFFu)     // global_addr[56:32]
                | (2u << 30);                              // type = 2 ("image")
        // D# group 1: data_size=4B; tensor 8192x8192, stride 8192; tile 8192x1.
        v8i g1;
        g1[0] = 0x20000;                 // [17:16] data_size=2 -> 4 bytes
        g1[1] = (int)(8192u << 16);      // [63:48]  tensor_dim0 lo16 = 8192
        g1[2] = (int)(8192u << 16);      // [79:64]  dim0 hi16=0 | [95:80] tensor_dim1 lo16
        g1[3] = (int)(8192u << 16);      // [111:96] dim1 hi16=0 | [127:112] tile_dim0=8192
        g1[4] = 1;                       // [143:128] tile_dim1=1 | [159:144] tile_dim2=0
        g1[5] = 8192;                    // [191:160] tensor_dim0_stride lo32
        g1[6] = 0;                       // stride hi16 | tensor_dim1_stride lo16
        g1[7] = 0;
        v4i z4 = {};
  #if __has_include(<hip/amd_detail/amd_gfx1250_TDM.h>)
        v8i z8 = {};
        __builtin_amdgcn_tensor_load_to_lds(g0, g1, z4, z4, z8, 0);
  #else
        __builtin_amdgcn_tensor_load_to_lds(g0, g1, z4, z4, 0);
  #endif
        __builtin_amdgcn_s_wait_tensorcnt(0);
    }
#else
    {
        const float4* gin  = (const float4*)(in + (size_t)row * 8192);
        float4*       lds4 = (float4*)lds;
        #pragma unroll
        for (int j = 0; j < 8; ++j) {
            lds4[j * 256 + tid] = gin[j * 256 + tid];
        }
    }
#endif
    __syncthreads();

    const int n    = lane & 15;   // N column within a tile / M row for A
    const int half = lane >> 4;   // lane half selects K/M sub-block

    // Constant A = H16 fragments, per ISA 16x4 f32 A layout:
    //   lane L, vgpr v -> M = L&15, K = 4*kb + 2*(L>>4) + v
    v2f Afrag[4];
    #pragma unroll
    for (int kb = 0; kb < 4; ++kb) {
        Afrag[kb].x = hsign(n, kb * 4 + half * 2 + 0);
        Afrag[kb].y = hsign(n, kb * 4 + half * 2 + 1);
    }

    // ================= Pass 1: axis a (K-stride 512) =================
    #pragma unroll
    for (int i = 0; i < 4; ++i) {
        const int t = wave * 4 + i;
        const int col = t * 16 + n;
        v8f c = {};
        #pragma unroll
        for (int kb = 0; kb < 4; ++kb) {
            const int k0 = kb * 4 + half * 2;
            v2f b;
            b.x = lds[(k0 + 0) * 512 + col];
            b.y = lds[(k0 + 1) * 512 + col];
            c = __builtin_amdgcn_wmma_f32_16x16x4_f32(false, Afrag[kb],
                                                      false, b,
                                                      (short)0, c, false, false);
        }
        #pragma unroll
        for (int v = 0; v < 8; ++v) {
            const int M = v + half * 8;
            lds[M * 512 + col] = c[v];
        }
    }
    __syncthreads();

    // ================= Pass 2: axis b (K-stride 32) =================
    #pragma unroll
    for (int i = 0; i < 4; ++i) {
        const int t = wave * 4 + i;
        const int a = t >> 1;
        const int r2 = (t & 1) * 16 + n;
        const int base = a * 512 + r2;
        v8f c = {};
        #pragma unroll
        for (int kb = 0; kb < 4; ++kb) {
            const int k0 = kb * 4 + half * 2;
            v2f b;
            b.x = lds[base + (k0 + 0) * 32];
            b.y = lds[base + (k0 + 1) * 32];
            c = __builtin_amdgcn_wmma_f32_16x16x4_f32(false, Afrag[kb],
                                                      false, b,
                                                      (short)0, c, false, false);
        }
        #pragma unroll
        for (int v = 0; v < 8; ++v) {
            const int M = v + half * 8;
            lds[base + M * 32] = c[v];
        }
    }
    __syncthreads();

    // ================= Pass 3: axis c (K-stride 2) =================
    #pragma unroll
    for (int i = 0; i < 4; ++i) {
        const int t = wave * 4 + i;
        const int a = t >> 1;
        const int bcol = (t & 1) * 8 + (n >> 1);
        const int d = n & 1;
        const int base = a * 512 + bcol * 32 + d;
        v8f c = {};
        #pragma unroll
        for (int kb = 0; kb < 4; ++kb) {
            const int k0 = kb * 4 + half * 2;
            v2f b;
            b.x = lds[base + (k0 + 0) * 2];
            b.y = lds[base + (k0 + 1) * 2];
            c = __builtin_amdgcn_wmma_f32_16x16x4_f32(false, Afrag[kb],
                                                      false, b,
                                                      (short)0, c, false, false);
        }
        #pragma unroll
        for (int v = 0; v < 8; ++v) {
            const int M = v + half * 8;
            lds[base + M * 2] = c[v];
        }
    }
    __syncthreads();

    // ====== Pass 4: axis d (H2) fused into non-temporal coalesced store =====
    {
        v2f*       gout = (v2f*)(out + (size_t)row * 8192);
        const v2f* lds2 = (const v2f*)lds;
        #pragma unroll
        for (int j = 0; j < 16; ++j) {
            const int p = j * 256 + tid;       // pair index 0..4095
            const v2f v = lds2[p];
            v2f r;
            r.x = v.x + v.y;
            r.y = v.x - v.y;
            __builtin_nontemporal_store(r, gout + p);   // write-once stream
        }
    }
}

extern "C" void kernel_launch(void* const* d_in, const int* in_sizes, int n_in,
                              void* d_out, int out_size, void* d_ws, size_t ws_size,
                              hipStream_t stream) {
    (void)n_in; (void)out_size; (void)d_ws; (void)ws_size;
    const float* x = (const float*)d_in[0];
    float* y = (float*)d_out;
    const int rows = in_sizes[0] >> 13;   // elements / 8192 (row length = 8192)
    fwht8192_wmma_kernel<<<dim3(rows), dim3(256), 0, stream>>>(x, y);
}